// CrossDenseLayer_16904991277331
// MI455X (gfx1250) — compile-verified
//
#include <hip/hip_runtime.h>
#include <hip/hip_bf16.h>
#include <stdint.h>

// Problem constants (from the reference)
#define B_   4
#define T_   32
#define N_   64
#define V_   8
#define F_   8
#define FI_  128
#define J_   16
#define R_   (T_ * N_)              // 2048 rows per GEMM
#define XROW_STRIDE   (V_ * F_ * FI_)   // 8192 floats between consecutive rows of x
#define OUTROW_STRIDE (V_ * F_ * J_)    // 1024 floats between consecutive rows of out
#define WVAR_STRIDE   (FI_ * F_ * J_)   // 16384 floats per variable in weights
#define WI_STRIDE     (F_ * J_)         // 128 floats between consecutive i in weights

#define WAVES_PER_BLOCK 4
#define LDS_XSTRIDE     132             // 128 + 4 pad -> conflict-free A-frag reads
#define LDS_X_PER_WAVE  (16 * LDS_XSTRIDE)                 // 2112 floats
#define LDS_W_OFF       (WAVES_PER_BLOCK * LDS_X_PER_WAVE) // 8448 floats
#define LDS_TOTAL       (LDS_W_OFF + FI_ * J_)             // 10496 floats = 41984 B

typedef float v2f __attribute__((ext_vector_type(2)));
typedef float v8f __attribute__((ext_vector_type(8)));

__global__ __launch_bounds__(WAVES_PER_BLOCK * 32)
void cross_dense_wmma_f32(const float* __restrict__ x,
                          const float* __restrict__ w,
                          const long long* __restrict__ emb_var,
                          float* __restrict__ out) {
  __shared__ float lds[LDS_TOTAL];

  const int tid  = threadIdx.x;
  const int lane = tid & 31;
  const int wave = tid >> 5;
  const int ml   = lane & 15;   // column / row-within-half index
  const int lh   = lane >> 4;   // lane-half

  // Grid: 256 GEMMs x 32 row-blocks; each block = 4 waves = 4 row tiles of 16.
  const int blk = blockIdx.x;
  const int g   = blk >> 5;     // GEMM id = (b*V + v)*F + m
  const int rb  = blk & 31;     // row-block within GEMM
  const int b   = g >> 6;       // / (V_*F_)
  const int v   = (g >> 3) & 7;
  const int m   = g & 7;

  const int var = (int)emb_var[b * V_ + v];
  const float* wbase = w + (size_t)var * WVAR_STRIDE + m * J_;   // + i*128 + j

  // ---- Stage W tile [128 x 16] into LDS (row stride 16). One i-row per thread.
  float* lw = lds + LDS_W_OFF;
  {
    const float* src = wbase + tid * WI_STRIDE;   // tid = i = 0..127
    float4 w0 = *(const float4*)(src + 0);
    float4 w1 = *(const float4*)(src + 4);
    float4 w2 = *(const float4*)(src + 8);
    float4 w3 = *(const float4*)(src + 12);
    float* dst = lw + tid * J_;
    *(float4*)(dst + 0)  = w0;
    *(float4*)(dst + 4)  = w1;
    *(float4*)(dst + 8)  = w2;
    *(float4*)(dst + 12) = w3;
  }

  // ---- Stage this wave's X tile [16 rows x 128] into LDS, coalesced b128 rows.
  const int r0 = (rb * WAVES_PER_BLOCK + wave) * 16;
  float* lx = lds + wave * LDS_X_PER_WAVE;
  const float* xg = x + ((size_t)b * R_) * XROW_STRIDE + (v * F_ + m) * FI_;
#pragma unroll
  for (int p = 0; p < 16; ++p) {
    float4 d = *(const float4*)(xg + (size_t)(r0 + p) * XROW_STRIDE + lane * 4);
    *(float4*)(lx + p * LDS_XSTRIDE + lane * 4) = d;
  }

  __syncthreads();

  // ---- K loop: 32 x V_WMMA_F32_16X16X4_F32, accumulate in C.
  v8f c = {0.f, 0.f, 0.f, 0.f, 0.f, 0.f, 0.f, 0.f};
#pragma unroll
  for (int kk = 0; kk < 32; ++kk) {
    const int k0 = kk * 4 + lh * 2;
    // A 16x4: lane-half selects K pair; lanes 0-15 are M=0..15.
    v2f a = *(const v2f*)(lx + ml * LDS_XSTRIDE + k0);     // ds_load_b64, bank-clean
    // B 4x16: lanes hold N=0..15; K pair per lane-half.
    v2f bf;
    bf.x = lw[(k0 + 0) * J_ + ml];
    bf.y = lw[(k0 + 1) * J_ + ml];
    c = __builtin_amdgcn_wmma_f32_16x16x4_f32(
        /*neg_a=*/false, a, /*neg_b=*/false, bf,
        /*c_mod=*/(short)0, c, /*reuse_a=*/false, /*reuse_b=*/false);
  }

  // ---- Store D: VGPR g holds (M = g + 8*lh, N = ml).
  float* og = out + ((size_t)b * R_ + r0 + lh * 8) * OUTROW_STRIDE
                  + (v * F_ + m) * J_ + ml;
#pragma unroll
  for (int gg = 0; gg < 8; ++gg) {
    og[(size_t)gg * OUTROW_STRIDE] = c[gg];
  }
}

extern "C" void kernel_launch(void* const* d_in, const int* in_sizes, int n_in,
                              void* d_out, int out_size, void* d_ws, size_t ws_size,
                              hipStream_t stream) {
  (void)in_sizes; (void)n_in; (void)out_size; (void)d_ws; (void)ws_size;
  const float*     x  = (const float*)d_in[0];
  const float*     w  = (const float*)d_in[1];
  const long long* ev = (const long long*)d_in[2];   // int64 per reference
  float*           o  = (float*)d_out;

  dim3 grid(B_ * V_ * F_ * 32);   // 256 GEMMs * 32 row-blocks = 8192 blocks
  dim3 block(WAVES_PER_BLOCK * 32);
  cross_dense_wmma_f32<<<grid, block, 0, stream>>>(x, w, ev, o);
}